// SSIM_5858335392012
// MI455X (gfx1250) — compile-verified
//
#include <hip/hip_runtime.h>
#include <cstdint>

// SSIM (3x3 box window, VALID) over NHWC [16,512,640,3] f32 -> [16,510,638,3] f32.
// Bandwidth-bound: ~210 MB min traffic -> ~9us at 23.3 TB/s.
// Strategy: 512 blocks (16 images x 32 row-bands of 16 output rows). 4-row LDS
// ring buffer filled by CDNA5 async global->LDS b128 DMA (ASYNCcnt), prefetch
// distance 1: while computing output row h (input rows h..h+2), row h+3 streams
// into LDS. s_wait_asynccnt 4 is safe because every thread issues >=4 chunk
// copies per row and ASYNCcnt retires in issue order.

#define N_IMG  16
#define H_IN   512
#define W_IN   640
#define C_CH   3
#define H_OUT  510
#define W_OUT  638
#define H_TILE 16
#define ROW_F  (W_IN * C_CH)      // 1920 floats per x/y row
#define CHUNKS_X 480              // 1920/4 float4 chunks
#define CHUNKS_M 160              // 640/4
#define CHUNKS_TOT (CHUNKS_X + CHUNKS_X + CHUNKS_M)  // 1120 -> 4..5 per thread

#define SSIM_C1 (0.01f * 0.01f)
#define SSIM_C2 (0.03f * 0.03f)

__device__ __forceinline__ void async_copy_b128(uint32_t lds_byte_off,
                                                unsigned long long gaddr) {
  // Per-lane DMA: copies 16 bytes from the global address in gaddr to the LDS byte offset in lds_byte_off; tracked by ASYNCcnt.
  asm volatile("global_load_async_to_lds_b128 %0, %1, off"
               :: "v"(lds_byte_off), "v"(gaddr) : "memory");
}

__device__ __forceinline__ void wait_async0() {
  asm volatile("s_wait_asynccnt 0x0" ::: "memory");
}

__device__ __forceinline__ void wait_async4() {
  // All but the newest row's copies have landed (each row = >=4 per-thread issues).
  asm volatile("s_wait_asynccnt 0x4" ::: "memory");
}

__global__ __launch_bounds__(256) void
ssim_5858335392012_kernel(const float* __restrict__ x,
                          const float* __restrict__ y,
                          const float* __restrict__ m,
                          float* __restrict__ out) {
  __shared__ __align__(16) float shX[4][ROW_F];   // 30 KB
  __shared__ __align__(16) float shY[4][ROW_F];   // 30 KB
  __shared__ __align__(16) float shM[4][W_IN];    // 10 KB  (71.7 KB total)

  const int t  = threadIdx.x;
  const int n  = blockIdx.y;
  const int h0 = blockIdx.x * H_TILE;                 // first output row of band
  const int rows_out = min(H_TILE, H_OUT - h0);
  const int rows_in  = rows_out + 2;

  const float inv9 = 1.0f / 9.0f;

  // Stage one absolute input row r into ring slot r&3 via async b128 DMA.
  auto stage = [&](int r) {
    const int slot = r & 3;
    const size_t rowbase = (size_t)(n * H_IN + r);
    const float* xrow = x + rowbase * ROW_F;
    const float* yrow = y + rowbase * ROW_F;
    const float* mrow = m + rowbase * W_IN;
    for (int cid = t; cid < CHUNKS_TOT; cid += 256) {
      uint32_t ldsoff;
      unsigned long long g;
      if (cid < CHUNKS_X) {
        ldsoff = (uint32_t)(uintptr_t)&shX[slot][cid * 4];
        g = (unsigned long long)(uintptr_t)(xrow + cid * 4);
      } else if (cid < 2 * CHUNKS_X) {
        const int c2 = cid - CHUNKS_X;
        ldsoff = (uint32_t)(uintptr_t)&shY[slot][c2 * 4];
        g = (unsigned long long)(uintptr_t)(yrow + c2 * 4);
      } else {
        const int c2 = cid - 2 * CHUNKS_X;
        ldsoff = (uint32_t)(uintptr_t)&shM[slot][c2 * 4];
        g = (unsigned long long)(uintptr_t)(mrow + c2 * 4);
      }
      async_copy_b128(ldsoff, g);
    }
  };

  // Fill the pipe: rows h0, h0+1, h0+2 in flight.
  stage(h0);
  stage(h0 + 1);
  stage(h0 + 2);

  for (int h = h0; h < h0 + rows_out; ++h) {
    const bool pf = (h + 3) < (h0 + rows_in);
    if (pf) {
      stage(h + 3);      // prefetch next input row while we compute this one
      wait_async4();     // rows h..h+2 landed; row h+3 may still be streaming
    } else {
      wait_async0();     // drain: everything landed
    }
    __syncthreads();     // all waves' copies for rows h..h+2 visible in LDS

    const int s0 = h & 3, s1 = (h + 1) & 3, s2 = (h + 2) & 3;

    for (int w = t; w < W_OUT; w += 256) {
      float ax[3]  = {0.f, 0.f, 0.f};
      float ay[3]  = {0.f, 0.f, 0.f};
      float axx[3] = {0.f, 0.f, 0.f};
      float ayy[3] = {0.f, 0.f, 0.f};
      float axy[3] = {0.f, 0.f, 0.f};
      float sm = 0.f;

      const int slots[3] = {s0, s1, s2};
      #pragma unroll
      for (int k = 0; k < 3; ++k) {
        const int sl = slots[k];
        const float* Xr = &shX[sl][w * 3];   // 9 contiguous floats: 3 taps x 3 ch
        const float* Yr = &shY[sl][w * 3];
        #pragma unroll
        for (int j = 0; j < 9; ++j) {
          const float xv = Xr[j];
          const float yv = Yr[j];
          const int c = j % 3;               // compile-time after unroll
          ax[c] += xv;
          ay[c] += yv;
          axx[c] = fmaf(xv, xv, axx[c]);
          ayy[c] = fmaf(yv, yv, ayy[c]);
          axy[c] = fmaf(xv, yv, axy[c]);
        }
        sm += shM[sl][w] + shM[sl][w + 1] + shM[sl][w + 2];
      }

      const float smv = sm * inv9;
      float* orow = out + ((size_t)(n * H_OUT + h) * W_OUT + w) * 3;
      #pragma unroll
      for (int c = 0; c < 3; ++c) {
        const float mux = ax[c] * inv9;
        const float muy = ay[c] * inv9;
        const float sigx  = axx[c] * inv9 - mux * mux;
        const float sigy  = ayy[c] * inv9 - muy * muy;
        const float sigxy = axy[c] * inv9 - mux * muy;
        const float nnum = (2.0f * mux * muy + SSIM_C1) * (2.0f * sigxy + SSIM_C2);
        const float dden = (mux * mux + muy * muy + SSIM_C1) * (sigx + sigy + SSIM_C2);
        float v = (1.0f - nnum / dden) * 0.5f;
        v = fminf(fmaxf(v, 0.0f), 1.0f);
        orow[c] = smv * v;
      }
    }
    __syncthreads();     // ring slot h&3 safe to overwrite next iteration
  }
}

extern "C" void kernel_launch(void* const* d_in, const int* in_sizes, int n_in,
                              void* d_out, int out_size, void* d_ws, size_t ws_size,
                              hipStream_t stream) {
  (void)in_sizes; (void)n_in; (void)out_size; (void)d_ws; (void)ws_size;
  const float* x = (const float*)d_in[0];
  const float* y = (const float*)d_in[1];
  const float* m = (const float*)d_in[2];
  float* out = (float*)d_out;

  dim3 grid((H_OUT + H_TILE - 1) / H_TILE, N_IMG);  // 32 x 16 = 512 blocks
  ssim_5858335392012_kernel<<<grid, 256, 0, stream>>>(x, y, m, out);
}